// CCHLoss_39951785787527
// MI455X (gfx1250) — compile-verified
//
#include <hip/hip_runtime.h>
#include <hip/hip_bf16.h>

typedef __attribute__((ext_vector_type(2))) float v2f;
typedef __attribute__((ext_vector_type(8))) float v8f;

#define BATCH 16
#define NPTS 2048
#define NTILES (NPTS / 16)        // 128 column tiles of 16 points
#define WAVES_PER_BLOCK 8         // 256 threads = 8 wave32

// For each of 16 query points (one row-tile, one wave), find min squared
// distance to all NPTS target points, using V_WMMA_F32_16X16X4_F32 for the
// -2*x.y cross terms (K=3 padded to 4).
__global__ __launch_bounds__(256) void nn_min_kernel(const float* __restrict__ X,
                                                     const float* __restrict__ Y,
                                                     float* __restrict__ out) {
  const int lane = threadIdx.x & 31;
  const int wave = threadIdx.x >> 5;
  const int tile = blockIdx.x * WAVES_PER_BLOCK + wave; // global row-tile id
  const int batch = tile / NTILES;
  const int itile = tile % NTILES;

  const int m  = lane & 15;   // point index within a 16-tile
  const int kh = lane >> 4;   // K-half: lanes 0-15 hold K=0,1; lanes 16-31 hold K=2,3

  const float* xb = X + (size_t)batch * NPTS * 3;
  const float* yb = Y + (size_t)batch * NPTS * 3;

  // A operand: 16x4 f32 (K=3, pad K=3 with zero). Lane layout per ISA 7.12.2.
  const float* xp = xb + (itile * 16 + m) * 3;
  const float x0 = xp[0], x1 = xp[1], x2c = xp[2];
  const float x2m = x0 * x0 + x1 * x1 + x2c * x2c; // |x|^2 for point m

  v2f a;
  a.x = kh ? x2c : x0;
  a.y = kh ? 0.0f : x1;

  // Broadcast |x|^2 for the 8 rows covered by this lane's C/D slots:
  // C/D slot (lane, vgpr v) holds row M = v + 8*(lane>=16).
  float x2r[8];
#pragma unroll
  for (int v = 0; v < 8; ++v)
    x2r[v] = __shfl(x2m, v + (kh << 3), 32);

  float acc[8];
#pragma unroll
  for (int v = 0; v < 8; ++v) acc[v] = 3.4e38f;

  // Per-lane walking pointer through the 16-point column tiles.
  const float* yp = yb + m * 3;

  for (int jt = 0; jt < NTILES; ++jt, yp += 48) {
    // Speculative prefetch of the next tile; OOB speculative prefetches are
    // silently dropped by hardware, so no guard needed (keeps the loop body
    // branchless and lets the unroller produce a single clean block).
    __builtin_prefetch(yp + 48, 0, 3); // global_prefetch_b8

    const float y0 = yp[0], y1 = yp[1], y2c = yp[2];
    const float y2 = y0 * y0 + y1 * y1 + y2c * y2c; // |y|^2 for column (lane&15)

    // B operand for Y^T (4x16): column-major B == A-style layout of Y rows.
    v2f b;
    b.x = kh ? y2c : y0;
    b.y = kh ? 0.0f : y1;

    v8f c = {};
    // D = X(16x4) * Y^T(4x16): 16x16 tile of dot products in one instruction.
    v8f d = __builtin_amdgcn_wmma_f32_16x16x4_f32(
        /*neg_a=*/false, a, /*neg_b=*/false, b,
        /*c_mod=*/(short)0, c, /*reuse_a=*/false, /*reuse_b=*/false);

    // dist = |x|^2 + (|y|^2 - 2*dot); defer the |x|^2 add to after the loop.
#pragma unroll
    for (int v = 0; v < 8; ++v)
      acc[v] = fminf(acc[v], y2 - 2.0f * d[v]);
  }

  // Min over the 16 columns held across each 16-lane half (wave32 xor-reduce).
#pragma unroll
  for (int v = 0; v < 8; ++v) {
    float t = acc[v];
    t = fminf(t, __shfl_xor(t, 1, 32));
    t = fminf(t, __shfl_xor(t, 2, 32));
    t = fminf(t, __shfl_xor(t, 4, 32));
    t = fminf(t, __shfl_xor(t, 8, 32));
    acc[v] = t;
  }

  // Lane 0 writes rows 0-7, lane 16 writes rows 8-15.
  if (m == 0) {
    float* o = out + (size_t)batch * NPTS + itile * 16 + (kh << 3);
#pragma unroll
    for (int v = 0; v < 8; ++v) o[v] = x2r[v] + acc[v];
  }
}

// loss = mean(cham_v * mask) + mean(cham_pred) + mean(pred_dw^2); loss_normals = 0
__global__ __launch_bounds__(256) void final_reduce_kernel(const float* __restrict__ cham_pred,
                                                           const float* __restrict__ cham_v,
                                                           const float* __restrict__ mask,
                                                           const float* __restrict__ pred_dw,
                                                           float* __restrict__ out) {
  __shared__ float s0[256], s1[256], s2[256];
  const int tid = threadIdx.x;
  float sp = 0.0f, sm = 0.0f, sw = 0.0f;

  for (int i = tid; i < BATCH * NPTS; i += 256) {
    sp += cham_pred[i];
    sm += cham_v[i] * mask[i];
  }
  for (int i = tid; i < BATCH * NPTS * 3; i += 256) {
    const float t = pred_dw[i];
    sw += t * t;
  }
  s0[tid] = sp; s1[tid] = sm; s2[tid] = sw;
  __syncthreads();
  for (int off = 128; off > 0; off >>= 1) {
    if (tid < off) {
      s0[tid] += s0[tid + off];
      s1[tid] += s1[tid + off];
      s2[tid] += s2[tid + off];
    }
    __syncthreads();
  }
  if (tid == 0) {
    const float inv_n  = 1.0f / (float)(BATCH * NPTS);
    const float inv_nw = 1.0f / (float)(BATCH * NPTS * 3);
    out[0] = s1[0] * inv_n + s0[0] * inv_n + s2[0] * inv_nw;
    out[1] = 0.0f;
  }
}

extern "C" void kernel_launch(void* const* d_in, const int* in_sizes, int n_in,
                              void* d_out, int out_size, void* d_ws, size_t ws_size,
                              hipStream_t stream) {
  const float* v       = (const float*)d_in[0]; // (B, P2, 3)
  const float* v_pred  = (const float*)d_in[1]; // (B, P1, 3)
  const float* mask    = (const float*)d_in[2]; // (4,4,2,32,32) -> flat (B, P2)
  const float* pred_dw = (const float*)d_in[3]; // (B, P1, 3)
  float* out = (float*)d_out;

  float* ws = (float*)d_ws;
  float* cham_pred = ws;                 // (B, P1) = 32768 floats
  float* cham_v    = ws + BATCH * NPTS;  // (B, P2) = 32768 floats

  const int blocks = BATCH * NTILES / WAVES_PER_BLOCK; // 256

  // cham_pred[b,i] = min_j ||v_pred[b,i] - v[b,j]||^2
  nn_min_kernel<<<blocks, 256, 0, stream>>>(v_pred, v, cham_pred);
  // cham_v[b,j] = min_i ||v[b,j] - v_pred[b,i]||^2  (same kernel, roles swapped)
  nn_min_kernel<<<blocks, 256, 0, stream>>>(v, v_pred, cham_v);

  final_reduce_kernel<<<1, 256, 0, stream>>>(cham_pred, cham_v, mask, pred_dw, out);
}